// AperiodicKNN_PyG_90666759618715
// MI455X (gfx1250) — compile-verified
//
#include <hip/hip_runtime.h>
#include <math.h>

// Aperiodic k-NN (N=16384, K=17) via f32 WMMA 16x16x4 dot-product tiles.
//
// d2[i][j] = |p_i|^2 + |p_j|^2 - 2 * (p_i . p_j); the 3-D dot products are
// computed 16x16 at a time with V_WMMA_F32_16X16X4_F32 (K=4, 4th comp = 0).
// One wave per block owns 32 center rows, scans all columns in 64-wide
// chunks, round-trips the dot tiles through LDS to transpose the WMMA D
// layout into one-row-per-lane, and keeps a sorted top-17 in registers.
//
// Round-2 change: all 4 column-point global_load_b96 per iteration are
// issued up front (batched MLP) instead of load->wait->use per tile.

#define KNN_K   17
#define RPB     32      // center rows per block (one wave)
#define CPI     64      // candidate columns per iteration (4 WMMA col-tiles)
#define LSTRIDE 68      // LDS row stride in floats (272B rows: 16B aligned)

typedef __attribute__((ext_vector_type(2))) float v2f;
typedef __attribute__((ext_vector_type(8))) float v8f;

__global__ __launch_bounds__(32)
void knn_wmma_kernel(const float* __restrict__ pos, int n,
                     int* __restrict__ out_src, int* __restrict__ out_dst,
                     float* __restrict__ out_dist)
{
    __shared__ __align__(16) float dotbuf[RPB * LSTRIDE];
    __shared__ __align__(16) float sqcol[CPI];

    const int  lane = threadIdx.x;      // 0..31
    const int  m    = lane & 15;
    const bool lo   = lane < 16;
    const int  rowbase = blockIdx.x * RPB;

    auto clampi = [n] __device__ (int i) { return i < n ? i : n - 1; };

    // ---- A tiles: rows [rowbase, rowbase+16) and [rowbase+16, rowbase+32) ----
    // ISA A layout (16x4 f32): lanes 0-15 hold (K0,K1)=(x,y) of row M=lane;
    // lanes 16-31 hold (K2,K3)=(z,0) of row M=lane-16.
    const float* pa0 = pos + 3 * clampi(rowbase + m);
    const float* pa1 = pos + 3 * clampi(rowbase + 16 + m);
    const float a0x = pa0[0], a0y = pa0[1], a0z = pa0[2];
    const float a1x = pa1[0], a1y = pa1[1], a1z = pa1[2];
    v2f a0, a1;
    a0[0] = lo ? a0x : a0z;  a0[1] = lo ? a0y : 0.0f;
    a1[0] = lo ? a1x : a1z;  a1[1] = lo ? a1y : 0.0f;

    // This lane's own center point (row gi = rowbase + lane).
    const int   gi = rowbase + lane;
    const float ox = lo ? a0x : a1x;
    const float oy = lo ? a0y : a1y;
    const float oz = lo ? a0z : a1z;
    const float sq_i = ox * ox + oy * oy + oz * oz;

    // Sorted top-K (ascending d2) kept entirely in VGPRs (static indexing).
    float kd[KNN_K]; int ki[KNN_K];
    #pragma unroll
    for (int t = 0; t < KNN_K; ++t) { kd[t] = INFINITY; ki[t] = 0; }

    for (int c0 = 0; c0 < n; c0 += CPI) {
        if (c0 + CPI < n) {   // warm L2/WGP$ for the next column chunk
            const char* nxt = (const char*)(pos + 3 * (c0 + CPI));
            __builtin_prefetch(nxt + lane * 32, 0, 1);
        }

        // ---- batch-issue all 4 column-point loads (one latency, not four) ----
        float cp[4][3];
        #pragma unroll
        for (int tc = 0; tc < 4; ++tc) {
            const int    col = clampi(c0 + tc * 16 + m);
            const float* pc  = pos + 3 * col;
            cp[tc][0] = pc[0]; cp[tc][1] = pc[1]; cp[tc][2] = pc[2];
        }

        // ---- compute 32x64 dot tile with 8 WMMAs, stage into LDS ----
        #pragma unroll
        for (int tc = 0; tc < 4; ++tc) {
            const float cx = cp[tc][0], cy = cp[tc][1], cz = cp[tc][2];
            sqcol[tc * 16 + m] = cx * cx + cy * cy + cz * cz; // both halves write same value

            // B layout (4x16) mirrors A per-lane: (x,y) low half, (z,0) high half.
            v2f b; b[0] = lo ? cx : cz; b[1] = lo ? cy : 0.0f;

            v8f acc = {};
            v8f d0 = __builtin_amdgcn_wmma_f32_16x16x4_f32(
                         false, a0, false, b, (short)0, acc, false, false);
            v8f d1 = __builtin_amdgcn_wmma_f32_16x16x4_f32(
                         false, a1, false, b, (short)0, acc, false, false);

            // D layout: VGPR k = row k (lanes 0-15) / row k+8 (lanes 16-31).
            const int cl = tc * 16 + m;
            const int rl = lo ? 0 : 8;
            #pragma unroll
            for (int k = 0; k < 8; ++k) {
                dotbuf[(rl + k)      * LSTRIDE + cl] = d0[k];
                dotbuf[(16 + rl + k) * LSTRIDE + cl] = d1[k];
            }
        }
        __syncthreads();   // single-wave WG: lowers to DS-count wait

        // ---- selection: lane L owns row L, scans 64 candidates from LDS ----
        const float* rowp = &dotbuf[lane * LSTRIDE];
        #pragma unroll 1
        for (int c = 0; c < CPI; c += 4) {
            const float4 dt = *(const float4*)(rowp + c);
            const float4 s4 = *(const float4*)(&sqcol[c]);
            const float dv[4] = { dt.x, dt.y, dt.z, dt.w };
            const float sv[4] = { s4.x, s4.y, s4.z, s4.w };
            #pragma unroll
            for (int u = 0; u < 4; ++u) {
                const int j = c0 + c + u;
                float d2 = sq_i + sv[u] - 2.0f * dv[u];
                d2 = (j == gi || j >= n) ? INFINITY : d2;
                if (d2 < kd[KNN_K - 1]) {            // rare after warm-up
                    float cd = d2; int cj = j;
                    #pragma unroll
                    for (int t = 0; t < KNN_K; ++t) {  // stable insert
                        if (cd < kd[t]) {
                            const float td = kd[t]; const int tj = ki[t];
                            kd[t] = cd; ki[t] = cj;
                            cd = td; cj = tj;
                        }
                    }
                }
            }
        }
        __syncthreads();
    }

    // ---- emit edges: src (neighbor), dst (center), sqrt distance ----
    if (gi < n) {
        const long base = (long)gi * KNN_K;
        #pragma unroll
        for (int t = 0; t < KNN_K; ++t) {
            out_src[base + t]  = ki[t];
            out_dst[base + t]  = gi;
            out_dist[base + t] = sqrtf(fmaxf(kd[t], 0.0f));
        }
    }
}

extern "C" void kernel_launch(void* const* d_in, const int* in_sizes, int n_in,
                              void* d_out, int out_size, void* d_ws, size_t ws_size,
                              hipStream_t stream) {
    const float* pos = (const float*)d_in[0];
    const int n = in_sizes[0] / 3;                 // pos is [N, 3] f32
    const long NK = (long)n * KNN_K;

    // d_out layout (tuple, flattened in return order, all 4-byte elems):
    //   [0,        NK)   edge_index row 0 (src = neighbor idx, int32)
    //   [NK,     2*NK)   edge_index row 1 (dst = center idx,   int32)
    //   [2*NK,   3*NK)   edge_dist (float32)
    int*   oi = (int*)d_out;
    float* od = (float*)d_out + 2 * NK;

    const int blocks = (n + RPB - 1) / RPB;
    knn_wmma_kernel<<<blocks, 32, 0, stream>>>(pos, n, oi, oi + NK, od);
}